// CoupledClustersLossV2_71502615544389
// MI455X (gfx1250) — compile-verified
//
#include <hip/hip_runtime.h>
#include <cstdint>

typedef __attribute__((ext_vector_type(2))) float v2f;
typedef __attribute__((ext_vector_type(8))) float v8f;

#define DDIM    2048
#define NCLS    256
#define NSAMP   32
#define ROWS    64                 // 32 pos + 32 neg rows per class
#define CHUNK   128                // K-chunk staged in LDS
#define NCHUNK  (DDIM / CHUNK)
#define LDSTR   (CHUNK + 4)        // stride mod 64 banks = 4 -> conflict-free col reads
#define NTILE   9                  // unique Gram tiles actually needed

// ---- gfx1250 async global->LDS (ASYNCcnt), via inline asm ----------------
// global_load_async_to_lds_b128 vDstLds, vAddr64, off
// vDstLds = 32-bit LDS byte address (low 32 bits of the flat pointer),
// vAddr64 = 64-bit global address in a VGPR pair.
__device__ __forceinline__ void async_load_b128(const void* gsrc, void* lds_dst) {
    uint32_t loff = (uint32_t)(uintptr_t)lds_dst;
    asm volatile("global_load_async_to_lds_b128 %0, %1, off"
                 :: "v"(loff), "v"(gsrc) : "memory");
}

template <int N>
__device__ __forceinline__ void wait_asynccnt() {
#if __has_builtin(__builtin_amdgcn_s_wait_asynccnt)
    __builtin_amdgcn_s_wait_asynccnt(N);
#else
    asm volatile("s_wait_asynccnt %0" :: "i"(N) : "memory");
#endif
}

// Issue the 64x128 f32 chunk for class slab `base`, chunk `ch`, into LDS buffer.
// 512 threads x 4 b128 ops -> 4 ASYNCcnt increments per wave, in-order completion.
__device__ __forceinline__ void issue_chunk_load(const float* __restrict__ base,
                                                 int ch, float* dst, int tid) {
    #pragma unroll
    for (int t = 0; t < 4; ++t) {
        int idx = tid + t * 512;          // 0..2047 float4 slots
        int row = idx >> 5;               // 0..63
        int cv  = idx & 31;               // float4 within row
        const float* src = base + (size_t)row * DDIM + ch * CHUNK + cv * 4;
        async_load_b128(src, &dst[row * LDSTR + cv * 4]);
        if (ch + 1 < NCHUNK)              // keep L2 warm for the chunk after
            __builtin_prefetch(src + CHUNK, 0, 1);
    }
}

// One 16x16 Gram tile K-slab: A rows rbase..rbase+15, B rows cbase..cbase+15.
// ISA 16x16x4 f32 A/B layout: VGPR0 = K 0/2 (lane halves), VGPR1 = K 1/3.
__device__ __forceinline__ void gram_tile(const float* buf, int rbase, int cbase,
                                          int m, int kh, v8f& acc) {
    const float* pa = &buf[(rbase + m) * LDSTR + 2 * kh];
    const float* pb = &buf[(cbase + m) * LDSTR + 2 * kh];
    #pragma unroll
    for (int kk = 0; kk < CHUNK; kk += 4) {
        v2f a; a.x = pa[kk]; a.y = pa[kk + 1];
        v2f b; b.x = pb[kk]; b.y = pb[kk + 1];
        acc = __builtin_amdgcn_wmma_f32_16x16x4_f32(
            false, a, false, b, (short)0, acc, false, false);
    }
}

// One workgroup per class; 16 waves; waves 0..8 each own one needed Gram tile.
__global__ __launch_bounds__(512)
void ccl_gram_kernel(const float* __restrict__ emb,
                     const float* __restrict__ marginp,
                     float* __restrict__ class_loss) {
    __shared__ float lds_e[2][ROWS * LDSTR];   // double-buffered staging (~66 KB)
    __shared__ float G[ROWS * ROWS];           // 16 KB Gram (only needed tiles written)
    __shared__ float dot_a[ROWS];
    __shared__ float diag[ROWS];

    const int tid  = threadIdx.x;
    const int cls  = blockIdx.x;
    const int lane = tid & 31;
    const int wv   = tid >> 5;              // 0..15 wave id
    const int m    = lane & 15;             // A row-in-tile == B col-in-tile
    const int kh   = lane >> 4;             // K-half selector (ISA A/B layout)

    // tile map: {(0,0),(1,0),(1,1),(2,0),(2,1),(3,0),(3,1),(2,2),(3,3)}
    int br = 0, bc = 0;
    switch (wv) {
        case 0: br = 0; bc = 0; break;
        case 1: br = 1; bc = 0; break;      // mirrored into (0,1) at scatter
        case 2: br = 1; bc = 1; break;
        case 3: br = 2; bc = 0; break;
        case 4: br = 2; bc = 1; break;
        case 5: br = 3; bc = 0; break;
        case 6: br = 3; bc = 1; break;
        case 7: br = 2; bc = 2; break;      // diag-only use
        case 8: br = 3; bc = 3; break;      // diag-only use
        default: break;
    }
    const bool active = (wv < NTILE);

    const float* base = emb + (size_t)cls * ROWS * DDIM;

    v8f acc = {};                           // 16x16 f32 C/D tile, 8 VGPRs

    // ---- software-pipelined async mem->LDS / WMMA loop ----
    issue_chunk_load(base, 0, lds_e[0], tid);               // prologue

    for (int ch = 0; ch < NCHUNK - 1; ++ch) {
        issue_chunk_load(base, ch + 1, lds_e[(ch + 1) & 1], tid);
        wait_asynccnt<4>();                 // oldest 4 (chunk ch) have landed
        __syncthreads();                    // all waves' chunk-ch data visible
        if (active) gram_tile(lds_e[ch & 1], br * 16, bc * 16, m, kh, acc);
        __syncthreads();                    // safe to overwrite this buffer later
    }
    wait_asynccnt<0>();
    __syncthreads();
    if (active) gram_tile(lds_e[(NCHUNK - 1) & 1], br * 16, bc * 16, m, kh, acc);

    // Scatter C tile into shared G. C layout: VGPR r -> M=r (lanes 0-15) / M=r+8.
    if (active) {
        int col = bc * 16 + m;
        int rb  = br * 16 + 8 * kh;
        #pragma unroll
        for (int r = 0; r < 8; ++r) {
            G[(rb + r) * ROWS + col] = acc[r];
            if (wv == 1)                    // mirror (1,0) -> (0,1) by symmetry
                G[col * ROWS + (rb + r)] = acc[r];
        }
    }
    __syncthreads();

    // dot_a[i] = x_i . anchor = (1/32) sum_{j<32} G[i][j] ; diag[i] = ||x_i||^2
    if (tid < ROWS) {
        float rs = 0.f;
        #pragma unroll 8
        for (int j = 0; j < NSAMP; ++j) rs += G[tid * ROWS + j];
        dot_a[tid] = rs * (1.0f / NSAMP);
        diag[tid]  = G[tid * ROWS + tid];
    }
    __syncthreads();

    if (tid == 0) {
        const float margin = marginp[0];
        float a2 = 0.f;                     // ||anchor||^2
        for (int i = 0; i < NSAMP; ++i) a2 += dot_a[i];
        a2 *= (1.0f / NSAMP);

        float nmin = 3.4e38f;               // min neg distance^2
        for (int j = 0; j < NSAMP; ++j) {
            float d2 = diag[NSAMP + j] - 2.f * dot_a[NSAMP + j] + a2;
            nmin = fminf(nmin, d2);
        }
        float an = sqrtf(fmaxf(nmin, 0.f));

        float loss = 0.f;
        for (int i = 0; i < NSAMP; ++i) {
            float d2 = diag[i] - 2.f * dot_a[i] + a2;
            float ap = sqrtf(fmaxf(d2, 0.f));
            float t  = ap - an + margin;
            if (t > 0.f) loss += t * t;
        }
        class_loss[cls] = loss;
    }
}

// Deterministic tree reduction of 256 per-class losses -> mean.
__global__ __launch_bounds__(256)
void ccl_reduce_kernel(const float* __restrict__ class_loss,
                       float* __restrict__ out) {
    __shared__ float s[NCLS];
    int tid = threadIdx.x;
    s[tid] = class_loss[tid];
    __syncthreads();
    #pragma unroll
    for (int stride = NCLS / 2; stride > 0; stride >>= 1) {
        if (tid < stride) s[tid] += s[tid + stride];
        __syncthreads();
    }
    if (tid == 0) out[0] = s[0] * (1.0f / NCLS);
}

extern "C" void kernel_launch(void* const* d_in, const int* in_sizes, int n_in,
                              void* d_out, int out_size, void* d_ws, size_t ws_size,
                              hipStream_t stream) {
    const float* emb    = (const float*)d_in[0];
    // d_in[1] = target (unused by the reference math)
    const float* margin = (const float*)d_in[2];
    float* cls_loss     = (float*)d_ws;     // 256 floats of scratch

    ccl_gram_kernel<<<NCLS, 512, 0, stream>>>(emb, margin, cls_loss);
    ccl_reduce_kernel<<<1, NCLS, 0, stream>>>(cls_loss, (float*)d_out);
}